// GATModel_75995151336045
// MI455X (gfx1250) — compile-verified
//
#include <hip/hip_runtime.h>
#include <stdint.h>

#define N_NODES 50000
#define IN_C 128
#define HID 8
#define HEADS 8
#define F1 64          /* HEADS*HID */
#define OUT_C 16
#define NEG_SLOPE 0.2f

typedef __attribute__((ext_vector_type(2))) float v2f;
typedef __attribute__((ext_vector_type(8))) float v8f;

// ---- order-preserving float <-> uint encoding for atomicMax on floats ----
__device__ __forceinline__ unsigned f2mono(float f) {
  unsigned u = __float_as_uint(f);
  return (u & 0x80000000u) ? ~u : (u | 0x80000000u);
}
__device__ __forceinline__ float mono2f(unsigned m) {
  unsigned u = (m & 0x80000000u) ? (m & 0x7fffffffu) : ~m;
  return __uint_as_float(u);
}

__global__ void k_zero(float* __restrict__ p, long long n) {
  long long g = (long long)blockIdx.x * blockDim.x + threadIdx.x;
  if (g < n) p[g] = 0.0f;
}

// ---------------- Layer-1 GEMM: h1[N,64] = x[N,128] @ W1[64,128]^T --------
// block = 128 threads (4 waves); wave w -> feature tile w (16 cols); blockIdx = node tile
__global__ void k_gemm1(const float* __restrict__ x, const float* __restrict__ W1,
                        float* __restrict__ h1) {
  const int wave = threadIdx.x >> 5;
  const int lane = threadIdx.x & 31;
  const int r    = lane & 15;    // row (A) / col (B,D) within tile
  const int hi   = lane >> 4;    // half-wave select
  const int node0 = blockIdx.x << 4;
  const int fbase = wave << 4;
  const float* xr = x  + (size_t)(node0 + r) * IN_C;
  const float* wr = W1 + (size_t)(fbase + r) * IN_C;   // B col f = fbase+r, B[k,f]=W1[f,k]
  v8f acc = {};
#pragma unroll
  for (int k = 0; k < IN_C; k += 4) {
    const int ko = k + 2 * hi;
    v2f a, b;
    a.x = xr[ko]; a.y = xr[ko + 1];
    b.x = wr[ko]; b.y = wr[ko + 1];
    acc = __builtin_amdgcn_wmma_f32_16x16x4_f32(false, a, false, b, (short)0, acc,
                                                false, false);
  }
  const int col = fbase + r;
#pragma unroll
  for (int vi = 0; vi < 8; ++vi) {
    const int m = vi + 8 * hi;
    h1[(size_t)(node0 + m) * F1 + col] = acc[vi];
  }
}

// ---------------- Layer-2 GEMM: h2[N,16] = hin[N,64] @ W2[16,64]^T --------
// block = 128 threads (4 waves); each wave handles one 16-node tile
__global__ void k_gemm2(const float* __restrict__ hin, const float* __restrict__ W2,
                        float* __restrict__ h2) {
  const int wave = threadIdx.x >> 5;
  const int lane = threadIdx.x & 31;
  const int tile = blockIdx.x * 4 + wave;
  if (tile >= (N_NODES / 16)) return;   // wave-uniform: EXEC all-ones for survivors
  const int r  = lane & 15;
  const int hi = lane >> 4;
  const int node0 = tile << 4;
  const float* xr = hin + (size_t)(node0 + r) * F1;
  const float* wr = W2  + (size_t)r * F1;
  v8f acc = {};
#pragma unroll
  for (int k = 0; k < F1; k += 4) {
    const int ko = k + 2 * hi;
    v2f a, b;
    a.x = xr[ko]; a.y = xr[ko + 1];
    b.x = wr[ko]; b.y = wr[ko + 1];
    acc = __builtin_amdgcn_wmma_f32_16x16x4_f32(false, a, false, b, (short)0, acc,
                                                false, false);
  }
#pragma unroll
  for (int vi = 0; vi < 8; ++vi) {
    const int m = vi + 8 * hi;
    h2[(size_t)(node0 + m) * OUT_C + r] = acc[vi];
  }
}

// ------------- per-node attention logits, layer 1 (H=8, C=8) -------------
__global__ void k_attn1(const float* __restrict__ h1, const float* __restrict__ aS,
                        const float* __restrict__ aD, float* __restrict__ as1,
                        float* __restrict__ ad1) {
  int g = blockIdx.x * blockDim.x + threadIdx.x;
  if (g >= N_NODES * HEADS) return;
  const int n = g >> 3, h = g & 7;
  const float* hv = h1 + (size_t)n * F1 + h * HID;
  float s = 0.f, d = 0.f;
#pragma unroll
  for (int c = 0; c < HID; ++c) {
    const float v = hv[c];
    s += v * aS[h * HID + c];
    d += v * aD[h * HID + c];
  }
  as1[g] = s; ad1[g] = d;
}

// ------------- per-node attention logits, layer 2 (H=1, C=16) ------------
__global__ void k_attn2(const float* __restrict__ h2, const float* __restrict__ aS,
                        const float* __restrict__ aD, float* __restrict__ as2,
                        float* __restrict__ ad2) {
  int n = blockIdx.x * blockDim.x + threadIdx.x;
  if (n >= N_NODES) return;
  const float* hv = h2 + (size_t)n * OUT_C;
  float s = 0.f, d = 0.f;
#pragma unroll
  for (int c = 0; c < OUT_C; ++c) {
    const float v = hv[c];
    s += v * aS[c];
    d += v * aD[c];
  }
  as2[n] = s; ad2[n] = d;
}

// ------------- edge pass 1: segment max of leakyReLU logits ---------------
template <int H>
__global__ void k_edge_max(const long long* __restrict__ src, const long long* __restrict__ dst,
                           const float* __restrict__ as, const float* __restrict__ ad,
                           unsigned* __restrict__ m, int E) {
  int e = blockIdx.x * blockDim.x + threadIdx.x;
  const int Etot = E + N_NODES;
  if (e >= Etot) return;
  const int s = (e < E) ? (int)src[e] : (e - E);
  const int d = (e < E) ? (int)dst[e] : (e - E);
#pragma unroll
  for (int h = 0; h < H; ++h) {
    float v = as[s * H + h] + ad[d * H + h];
    v = v > 0.f ? v : NEG_SLOPE * v;
    atomicMax(&m[d * H + h], f2mono(v));
  }
}

// ------------- edge pass 2: segment sum of exp(e - max) -------------------
template <int H>
__global__ void k_edge_sum(const long long* __restrict__ src, const long long* __restrict__ dst,
                           const float* __restrict__ as, const float* __restrict__ ad,
                           const unsigned* __restrict__ m, float* __restrict__ den, int E) {
  int e = blockIdx.x * blockDim.x + threadIdx.x;
  const int Etot = E + N_NODES;
  if (e >= Etot) return;
  const int s = (e < E) ? (int)src[e] : (e - E);
  const int d = (e < E) ? (int)dst[e] : (e - E);
#pragma unroll
  for (int h = 0; h < H; ++h) {
    float v = as[s * H + h] + ad[d * H + h];
    v = v > 0.f ? v : NEG_SLOPE * v;
    atomicAdd(&den[d * H + h], __expf(v - mono2f(m[d * H + h])));
  }
}

// ------------- edge pass 3 (layer 1): scatter alpha*h1[src] ---------------
// one 64-thread block (2 waves) per edge; thread t -> head t/8, channel t%8
__global__ void k_scatter1(const long long* __restrict__ src, const long long* __restrict__ dst,
                           const float* __restrict__ as, const float* __restrict__ ad,
                           const unsigned* __restrict__ m, const float* __restrict__ den,
                           const float* __restrict__ h1, float* __restrict__ acc, int E) {
  const int e = blockIdx.x;
  const int t = threadIdx.x;          // 0..63
  const int s = (e < E) ? (int)src[e] : (e - E);
  const int d = (e < E) ? (int)dst[e] : (e - E);
  const int h = t >> 3;
  float v = as[s * HEADS + h] + ad[d * HEADS + h];
  v = v > 0.f ? v : NEG_SLOPE * v;
  const float alpha = __expf(v - mono2f(m[d * HEADS + h])) / (den[d * HEADS + h] + 1e-16f);
  atomicAdd(&acc[(size_t)d * F1 + t], alpha * h1[(size_t)s * F1 + t]);
}

// ------------- edge pass 3 (layer 2): thread per edge, 16 features --------
__global__ void k_scatter2(const long long* __restrict__ src, const long long* __restrict__ dst,
                           const float* __restrict__ as, const float* __restrict__ ad,
                           const unsigned* __restrict__ m, const float* __restrict__ den,
                           const float* __restrict__ h2, float* __restrict__ out, int E) {
  int e = blockIdx.x * blockDim.x + threadIdx.x;
  const int Etot = E + N_NODES;
  if (e >= Etot) return;
  const int s = (e < E) ? (int)src[e] : (e - E);
  const int d = (e < E) ? (int)dst[e] : (e - E);
  float v = as[s] + ad[d];
  v = v > 0.f ? v : NEG_SLOPE * v;
  const float alpha = __expf(v - mono2f(m[d])) / (den[d] + 1e-16f);
  const float4* hs = (const float4*)(h2 + (size_t)s * OUT_C);
#pragma unroll
  for (int q = 0; q < 4; ++q) {
    const float4 hv = hs[q];
    atomicAdd(&out[(size_t)d * OUT_C + q * 4 + 0], alpha * hv.x);
    atomicAdd(&out[(size_t)d * OUT_C + q * 4 + 1], alpha * hv.y);
    atomicAdd(&out[(size_t)d * OUT_C + q * 4 + 2], alpha * hv.z);
    atomicAdd(&out[(size_t)d * OUT_C + q * 4 + 3], alpha * hv.w);
  }
}

// ------------- finalize layer 1: +bias, ELU (in place) --------------------
__global__ void k_elu(float* __restrict__ acc, const float* __restrict__ b1) {
  int g = blockIdx.x * blockDim.x + threadIdx.x;
  if (g >= N_NODES * F1) return;
  const float v = acc[g] + b1[g & (F1 - 1)];
  acc[g] = v > 0.f ? v : (__expf(v) - 1.f);
}

// ------------- finalize layer 2: +bias ------------------------------------
__global__ void k_bias2(float* __restrict__ out, const float* __restrict__ b2) {
  int g = blockIdx.x * blockDim.x + threadIdx.x;
  if (g >= N_NODES * OUT_C) return;
  out[g] += b2[g & (OUT_C - 1)];
}

extern "C" void kernel_launch(void* const* d_in, const int* in_sizes, int n_in,
                              void* d_out, int out_size, void* d_ws, size_t ws_size,
                              hipStream_t stream) {
  (void)n_in; (void)out_size; (void)ws_size;
  const float*     x    = (const float*)d_in[0];
  const long long* ei   = (const long long*)d_in[1];  // int64 [2,E]
  const float*     W1   = (const float*)d_in[2];
  const float*     aS1  = (const float*)d_in[3];
  const float*     aD1  = (const float*)d_in[4];
  const float*     b1   = (const float*)d_in[5];
  const float*     W2   = (const float*)d_in[6];
  const float*     aS2  = (const float*)d_in[7];
  const float*     aD2  = (const float*)d_in[8];
  const float*     b2   = (const float*)d_in[9];

  const int E    = in_sizes[1] / 2;
  const int Etot = E + N_NODES;
  const long long* srcI = ei;
  const long long* dstI = ei + E;

  // -------- workspace layout (all 4-byte elems) --------
  float* ws  = (float*)d_ws;
  float* h1  = ws;                                   // N*64
  float* as1 = h1  + (size_t)N_NODES * F1;           // N*8
  float* ad1 = as1 + (size_t)N_NODES * HEADS;        // N*8
  float* h2  = ad1 + (size_t)N_NODES * HEADS;        // N*16
  float* as2 = h2  + (size_t)N_NODES * OUT_C;        // N
  float* ad2 = as2 + N_NODES;                        // N
  // zero region (re-initialized every call):
  float*    acc1 = ad2 + N_NODES;                    // N*64
  unsigned* m1   = (unsigned*)(acc1 + (size_t)N_NODES * F1);   // N*8
  float*    den1 = (float*)(m1 + (size_t)N_NODES * HEADS);     // N*8
  unsigned* m2   = (unsigned*)(den1 + (size_t)N_NODES * HEADS);// N
  float*    den2 = (float*)(m2 + N_NODES);                     // N
  const long long zeroCount = (long long)N_NODES * F1 + 2LL * N_NODES * HEADS + 2LL * N_NODES;

  float* out = (float*)d_out;                        // N*16

  // 0) zero accumulators / softmax stats / output
  {
    long long n = zeroCount;
    k_zero<<<(unsigned)((n + 255) / 256), 256, 0, stream>>>(acc1, n);
    long long no = (long long)N_NODES * OUT_C;
    k_zero<<<(unsigned)((no + 255) / 256), 256, 0, stream>>>(out, no);
  }

  // 1) layer-1 projection (WMMA) + logits
  k_gemm1<<<N_NODES / 16, 128, 0, stream>>>(x, W1, h1);
  k_attn1<<<(N_NODES * HEADS + 255) / 256, 256, 0, stream>>>(h1, aS1, aD1, as1, ad1);

  // 2) layer-1 segment softmax + scatter
  k_edge_max<HEADS><<<(Etot + 255) / 256, 256, 0, stream>>>(srcI, dstI, as1, ad1, m1, E);
  k_edge_sum<HEADS><<<(Etot + 255) / 256, 256, 0, stream>>>(srcI, dstI, as1, ad1, m1, den1, E);
  k_scatter1<<<Etot, 64, 0, stream>>>(srcI, dstI, as1, ad1, m1, den1, h1, acc1, E);
  k_elu<<<(N_NODES * F1 + 255) / 256, 256, 0, stream>>>(acc1, b1);

  // 3) layer-2 projection (WMMA) + logits
  k_gemm2<<<(N_NODES / 16 + 3) / 4, 128, 0, stream>>>(acc1, W2, h2);
  k_attn2<<<(N_NODES + 255) / 256, 256, 0, stream>>>(h2, aS2, aD2, as2, ad2);

  // 4) layer-2 segment softmax + scatter into output
  k_edge_max<1><<<(Etot + 255) / 256, 256, 0, stream>>>(srcI, dstI, as2, ad2, m2, E);
  k_edge_sum<1><<<(Etot + 255) / 256, 256, 0, stream>>>(srcI, dstI, as2, ad2, m2, den2, E);
  k_scatter2<<<(Etot + 255) / 256, 256, 0, stream>>>(srcI, dstI, as2, ad2, m2, den2, h2, out, E);
  k_bias2<<<(N_NODES * OUT_C + 255) / 256, 256, 0, stream>>>(out, b2);
}